// CostSensitiveCrossEntropyLossN_old_90142773609052
// MI455X (gfx1250) — compile-verified
//
#include <hip/hip_runtime.h>
#include <hip/hip_bf16.h>

// ---------------------------------------------------------------------------
// CostSensitiveCrossEntropyLoss for MI455X (gfx1250, wave32)
//
// Pipeline (all on `stream`):
//   K0 zero_u32      : clear confusion-matrix counts + row counts (integer)
//   K1 rowsum_in     : row sums of the *input* cost_matrix (async global->LDS
//                      staging via gfx1250 GLOBAL_LOAD_ASYNC_TO_LDS_B128)
//   K2 row_kernel    : 1 wave32 per row of outputs[65536,1000]:
//                        - 8x float4 loads/lane (global_load_b128, row in regs)
//                        - wave max+argmax, sum exp(x-m), gather x[target]
//                        - u32 atomics into cm counts / row counts
//   K3 weight_kernel : per-sample logp * beta*cm/max(1,rowsum), fixed-order
//                      block tree reduction -> 256 partials
//   K4 final_kernel  : deterministic 256-wide reduction, out = -sum/B
// ---------------------------------------------------------------------------

#define BETA_F 3.0f
#define NBLK_W 256           // weight-kernel blocks == number of partials
#define C_MAX_FAST 1024      // fast register path supports C <= 1024, C%4==0

typedef int v4i __attribute__((ext_vector_type(4)));
typedef __attribute__((address_space(1))) v4i* gv4i_ptr;
typedef __attribute__((address_space(3))) v4i* lv4i_ptr;

#if __has_builtin(__builtin_amdgcn_global_load_async_to_lds_b128)
#define HAVE_ASYNC_LDS 1
#else
#define HAVE_ASYNC_LDS 0
#endif

__device__ __forceinline__ void async_wait0() {
#if __has_builtin(__builtin_amdgcn_s_wait_asynccnt)
  __builtin_amdgcn_s_wait_asynccnt(0);
#else
  asm volatile("s_wait_asynccnt 0" ::: "memory");
#endif
  asm volatile("" ::: "memory");
}

__device__ __forceinline__ float wave_sum(float v) {
#pragma unroll
  for (int o = 16; o; o >>= 1) v += __shfl_xor(v, o, 32);
  return v;
}

// ---------------------------------------------------------------- K0: memset
__global__ __launch_bounds__(256) void zero_u32(unsigned int* p, long long n) {
  long long i = (long long)blockIdx.x * blockDim.x + threadIdx.x;
  long long stride = (long long)gridDim.x * blockDim.x;
  for (; i < n; i += stride) p[i] = 0u;
}

// ------------------------------------------------- K1: input-cm row sums
// One wave32 per row of cost_matrix[C,C]. Uses the CDNA5 async global->LDS
// DMA path (ASYNCcnt) when available; falls back to direct loads.
__global__ __launch_bounds__(256) void rowsum_in_kernel(
    const float* __restrict__ cm_in, float* __restrict__ rowsum, int C) {
#if HAVE_ASYNC_LDS
  __shared__ float lds[8 * C_MAX_FAST];
#endif
  const int wave = threadIdx.x >> 5;
  const int lane = threadIdx.x & 31;
  const int row = blockIdx.x * (blockDim.x >> 5) + wave;
  if (row >= C) return;
  const float* rp = cm_in + (long long)row * C;
  float s = 0.0f;
#if HAVE_ASYNC_LDS
  if (C <= C_MAX_FAST && (C & 3) == 0) {
    float* lrow = &lds[wave * C_MAX_FAST];
    const int nv4 = C >> 2;
    for (int idx = lane; idx < nv4; idx += 32) {
      __builtin_amdgcn_global_load_async_to_lds_b128(
          (gv4i_ptr)(rp + 4 * idx),
          (lv4i_ptr)(lrow + 4 * idx),
          0, 0);
    }
    async_wait0();
    for (int c = lane; c < C; c += 32) s += lrow[c];
  } else
#endif
  {
    for (int c = lane; c < C; c += 32) s += rp[c];
  }
  s = wave_sum(s);
  if (lane == 0) rowsum[row] = s;
}

// ------------------------------------------------- K2: main per-row kernel
__global__ __launch_bounds__(256, 2) void row_kernel(
    const float* __restrict__ outp, const long long* __restrict__ tgt,
    unsigned int* __restrict__ cm_cnt, unsigned int* __restrict__ row_cnt,
    int* __restrict__ pred, float* __restrict__ logp, int B, int C) {
  const int wid  = (int)((blockIdx.x * blockDim.x + threadIdx.x) >> 5);
  const int lane = threadIdx.x & 31;
  if (wid >= B) return;

  const int t = (int)tgt[wid];
  const float NEG = -__builtin_inff();

  float m = NEG;       // lane-local max
  int   mi = 0x7fffffff;
  float s = 0.0f;      // lane-local sum exp(x - m_global)
  float vt = 0.0f;     // lane-local x[target] (one lane nonzero)

  if (C <= C_MAX_FAST && (C & 3) == 0) {
    // ---- fast path: whole row in registers, single global read ----
    const float4* row = (const float4*)(outp + (long long)wid * C);
    const int nv4 = C >> 2;
    float4 v[8];
#pragma unroll
    for (int k = 0; k < 8; ++k) {
      const int idx = lane + (k << 5);
      if (idx < nv4) v[k] = row[idx];
      else           v[k] = make_float4(NEG, NEG, NEG, NEG);
    }
#pragma unroll
    for (int k = 0; k < 8; ++k) {
      const int base = (lane + (k << 5)) << 2;
      if (v[k].x > m) { m = v[k].x; mi = base + 0; }
      if (v[k].y > m) { m = v[k].y; mi = base + 1; }
      if (v[k].z > m) { m = v[k].z; mi = base + 2; }
      if (v[k].w > m) { m = v[k].w; mi = base + 3; }
    }
    // wave argmax (first occurrence on ties)
#pragma unroll
    for (int o = 16; o; o >>= 1) {
      const float om = __shfl_xor(m, o, 32);
      const int   oi = __shfl_xor(mi, o, 32);
      if (om > m || (om == m && oi < mi)) { m = om; mi = oi; }
    }
#pragma unroll
    for (int k = 0; k < 8; ++k) {
      const int base = (lane + (k << 5)) << 2;
      s += __expf(v[k].x - m) + __expf(v[k].y - m) +
           __expf(v[k].z - m) + __expf(v[k].w - m);
      const int rel = t - base;
      if (rel >= 0 && rel < 4)
        vt = (rel == 0) ? v[k].x : (rel == 1) ? v[k].y
           : (rel == 2) ? v[k].z : v[k].w;
    }
  } else {
    // ---- generic fallback: two passes ----
    const float* rp = outp + (long long)wid * C;
    for (int c = lane; c < C; c += 32) {
      const float x = rp[c];
      if (x > m) { m = x; mi = c; }
    }
#pragma unroll
    for (int o = 16; o; o >>= 1) {
      const float om = __shfl_xor(m, o, 32);
      const int   oi = __shfl_xor(mi, o, 32);
      if (om > m || (om == m && oi < mi)) { m = om; mi = oi; }
    }
    for (int c = lane; c < C; c += 32) {
      const float x = rp[c];
      s += __expf(x - m);
      if (c == t) vt = x;
    }
  }

  s  = wave_sum(s);
  vt = wave_sum(vt);
  const float logZ = m + logf(s);

  if (lane == 0) {
    pred[wid] = mi;
    logp[wid] = vt - logZ;
    atomicAdd(&cm_cnt[(long long)t * C + mi], 1u);
    atomicAdd(&row_cnt[t], 1u);
  }
}

// ------------------------------------------------- K3: per-sample weighting
__global__ __launch_bounds__(256) void weight_kernel(
    const long long* __restrict__ tgt, const int* __restrict__ pred,
    const float* __restrict__ logp, const float* __restrict__ cm_in,
    const unsigned int* __restrict__ cm_cnt,
    const float* __restrict__ rowsum_in, const unsigned int* __restrict__ row_cnt,
    float* __restrict__ partials, int B, int C) {
  __shared__ float sm[256];
  float acc = 0.0f;
  for (long long b = (long long)blockIdx.x * blockDim.x + threadIdx.x; b < B;
       b += (long long)gridDim.x * blockDim.x) {
    const int t = (int)tgt[b];
    const int p = pred[b];
    const float cm = cm_in[(long long)t * C + p] +
                     (float)cm_cnt[(long long)t * C + p];
    const float rs = rowsum_in[t] + (float)row_cnt[t];
    acc += logp[b] * cm * (BETA_F / fmaxf(1.0f, rs));
  }
  sm[threadIdx.x] = acc;
  __syncthreads();
#pragma unroll
  for (int sft = 128; sft > 0; sft >>= 1) {
    if ((int)threadIdx.x < sft) sm[threadIdx.x] += sm[threadIdx.x + sft];
    __syncthreads();
  }
  if (threadIdx.x == 0) partials[blockIdx.x] = sm[0];
}

// ------------------------------------------------- K4: final scalar
__global__ __launch_bounds__(256) void final_kernel(
    const float* __restrict__ partials, float* __restrict__ out, int nblk,
    float invB) {
  __shared__ float sm[256];
  sm[threadIdx.x] = ((int)threadIdx.x < nblk) ? partials[threadIdx.x] : 0.0f;
  __syncthreads();
#pragma unroll
  for (int sft = 128; sft > 0; sft >>= 1) {
    if ((int)threadIdx.x < sft) sm[threadIdx.x] += sm[threadIdx.x + sft];
    __syncthreads();
  }
  if (threadIdx.x == 0) out[0] = -sm[0] * invB;
}

// ---------------------------------------------------------------------------
extern "C" void kernel_launch(void* const* d_in, const int* in_sizes, int n_in,
                              void* d_out, int out_size, void* d_ws,
                              size_t ws_size, hipStream_t stream) {
  const float*     outputs = (const float*)d_in[0];
  const long long* targets = (const long long*)d_in[1];
  const float*     cm_in   = (const float*)d_in[2];

  const int B = in_sizes[1];               // 65536
  const int C = in_sizes[0] / B;           // 1000

  // Workspace layout (16B-aligned slabs). cm_cnt and row_cnt are contiguous
  // so one memset launch clears both.
  char* ws = (char*)d_ws;
  auto align16 = [](size_t x) { return (x + 15) & ~(size_t)15; };
  size_t off = 0;
  unsigned int* cm_cnt = (unsigned int*)(ws + off);
  off += (size_t)C * C * sizeof(unsigned int);            // stays 4B-aligned
  unsigned int* row_cnt = (unsigned int*)(ws + off);
  off = align16(off + (size_t)C * sizeof(unsigned int));
  float* rowsum_in = (float*)(ws + off);
  off = align16(off + (size_t)C * sizeof(float));
  int* pred = (int*)(ws + off);
  off = align16(off + (size_t)B * sizeof(int));
  float* logp = (float*)(ws + off);
  off = align16(off + (size_t)B * sizeof(float));
  float* partials = (float*)(ws + off);
  off = align16(off + (size_t)NBLK_W * sizeof(float));
  (void)ws_size; (void)n_in; (void)out_size;

  // K0: clear counts (cm_cnt followed immediately by row_cnt)
  {
    const long long n = (long long)C * C + C;
    const int blocks = (int)((n + 255) / 256) < 2048 ? (int)((n + 255) / 256)
                                                     : 2048;
    zero_u32<<<blocks, 256, 0, stream>>>(cm_cnt, n);
  }
  // K1: input cost-matrix row sums (async global->LDS path on gfx1250)
  {
    const int waves_per_blk = 8;
    const int blocks = (C + waves_per_blk - 1) / waves_per_blk;
    rowsum_in_kernel<<<blocks, 256, 0, stream>>>(cm_in, rowsum_in, C);
  }
  // K2: main per-row pass (1 wave32 per row, 8 waves per block)
  {
    const int rows_per_blk = 8;
    const int blocks = (B + rows_per_blk - 1) / rows_per_blk;
    row_kernel<<<blocks, 256, 0, stream>>>(outputs, targets, cm_cnt, row_cnt,
                                           pred, logp, B, C);
  }
  // K3: weighted per-sample terms -> NBLK_W partials
  weight_kernel<<<NBLK_W, 256, 0, stream>>>(targets, pred, logp, cm_in, cm_cnt,
                                            rowsum_in, row_cnt, partials, B, C);
  // K4: final deterministic reduction
  final_kernel<<<1, 256, 0, stream>>>(partials, (float*)d_out, NBLK_W,
                                      1.0f / (float)B);
}